// MultiresolutionHashEncoding_77841987272738
// MI455X (gfx1250) — compile-verified
//
#include <hip/hip_runtime.h>
#include <cmath>

// ---------------------------------------------------------------------------
// Multiresolution hash encoding (Instant-NGP style), 2D input, 16 levels,
// feature dim 2, table 2^19 entries/level.
//
// Bound analysis (MI455X): ~0.5 GFLOP vs 136 MB streaming + 512 MB of random
// 8B gathers into 64 MB of tables (L2-resident; L2 = 192 MB). => pure memory /
// gather-latency problem. Strategy:
//   * one thread per point, 16-level loop fully unrolled -> up to 64
//     independent global_load_b64 gathers in flight per lane (MLP)
//   * NT stores for the 128 MB output + NT load for x, so the streaming
//     traffic does not evict the L2-resident tables (gfx1250 TH hints)
//   * wave32, 256-thread blocks, VGPR budget capped for occupancy
// ---------------------------------------------------------------------------

#define ENC_LEVELS     16
#define ENC_TABLE_SIZE 524288u
#define ENC_MASK       (ENC_TABLE_SIZE - 1u)
#define ENC_PRIME_Y    2654435761u

typedef float v2f __attribute__((ext_vector_type(2)));
typedef float v4f __attribute__((ext_vector_type(4)));

struct ResTable { float r[ENC_LEVELS]; };

__global__ __launch_bounds__(256, 6)
void hashenc2d_kernel(const float* __restrict__ x,
                      const float* __restrict__ tables,
                      float* __restrict__ out,
                      int npts,
                      ResTable res)
{
    const int n = blockIdx.x * blockDim.x + threadIdx.x;
    if (n >= npts) return;

    // Streaming, read-once input: non-temporal b64 load.
    const v2f p = __builtin_nontemporal_load((const v2f*)x + n);

    float acc0[ENC_LEVELS];   // feature 0 per level
    float acc1[ENC_LEVELS];   // feature 1 per level

#pragma unroll
    for (int l = 0; l < ENC_LEVELS; ++l) {
        const float r  = res.r[l];
        const float sx = p.x * r;
        const float sy = p.y * r;
        const float gx = floorf(sx);
        const float gy = floorf(sy);
        const float fx = sx - gx;           // exact (Sterbenz)
        const float fy = sy - gy;

        const unsigned cx  = (unsigned)(int)gx;
        const unsigned cy  = (unsigned)(int)gy;
        const unsigned cyp = cy * ENC_PRIME_Y;       // prime_x == 1
        const unsigned cyq = cyp + ENC_PRIME_Y;      // (cy+1)*prime
        const unsigned h00 = (cx        ^ cyp) & ENC_MASK;
        const unsigned h10 = ((cx + 1u) ^ cyp) & ENC_MASK;
        const unsigned h01 = (cx        ^ cyq) & ENC_MASK;
        const unsigned h11 = ((cx + 1u) ^ cyq) & ENC_MASK;

        // Per-level table base: +l*4MB scalar offset. Tables stay RT-cached
        // (heavily reused, L2-resident).
        const v2f* __restrict__ tl = (const v2f*)tables + (size_t)l * ENC_TABLE_SIZE;
        const v2f f00 = tl[h00];     // global_load_b64 gathers
        const v2f f10 = tl[h10];
        const v2f f01 = tl[h01];
        const v2f f11 = tl[h11];

        // Reference weights: w_c = |s - corner_x| * |s - corner_y|
        const float w00 = fx * fy;
        const float w10 = (1.0f - fx) * fy;
        const float w01 = fx * (1.0f - fy);
        const float w11 = (1.0f - fx) * (1.0f - fy);

        acc0[l] = w00 * f00.x + w10 * f10.x + w01 * f01.x + w11 * f11.x;
        acc1[l] = w00 * f00.y + w10 * f10.y + w01 * f01.y + w11 * f11.y;
    }

    // out[n][f*16 + l]: 32 contiguous floats per point -> 8 NT b128 stores.
    // NT keeps the 128 MB output stream from evicting the tables in L2.
    float* o = out + (size_t)n * (2 * ENC_LEVELS);
#pragma unroll
    for (int j = 0; j < 4; ++j) {
        v4f v = { acc0[4 * j], acc0[4 * j + 1], acc0[4 * j + 2], acc0[4 * j + 3] };
        __builtin_nontemporal_store(v, (v4f*)o + j);
    }
#pragma unroll
    for (int j = 0; j < 4; ++j) {
        v4f v = { acc1[4 * j], acc1[4 * j + 1], acc1[4 * j + 2], acc1[4 * j + 3] };
        __builtin_nontemporal_store(v, (v4f*)(o + ENC_LEVELS) + j);
    }
}

extern "C" void kernel_launch(void* const* d_in, const int* in_sizes, int n_in,
                              void* d_out, int out_size, void* d_ws, size_t ws_size,
                              hipStream_t stream)
{
    const float* x      = (const float*)d_in[0];   // (N, 2) f32
    const float* tables = (const float*)d_in[1];   // (16, 524288, 2) f32
    float* out          = (float*)d_out;           // (N, 32) f32

    const int npts = in_sizes[0] / 2;

    // Reproduce numpy's resolution table bit-exactly (double math, then floor):
    // _b = exp((log(512) - log(16)) / 15); res_i = floor(16 * _b**i)
    ResTable rt;
    const double b = exp((log(512.0) - log(16.0)) / 15.0);
    for (int i = 0; i < ENC_LEVELS; ++i)
        rt.r[i] = (float)floor(16.0 * pow(b, (double)i));

    const int block = 256;
    const int grid  = (npts + block - 1) / block;
    hipLaunchKernelGGL(hashenc2d_kernel, dim3(grid), dim3(block), 0, stream,
                       x, tables, out, npts, rt);
}